// ModAttn_31190052503594
// MI455X (gfx1250) — compile-verified
//
#include <hip/hip_runtime.h>
#include <hip/hip_bf16.h>

// B=4, T=2048, F=4, E=256 -> FE=1024, DC=64, F*DC=256
#define Bn   4
#define Tn   2048
#define FE   1024
#define CDIM 256
#define LN_EPS 1e-5f

typedef __attribute__((ext_vector_type(2))) float v2f;
typedef __attribute__((ext_vector_type(8))) float v8f;

// ---------------------------------------------------------------------------
// Stage 1a: partial sums of x over t-chunks.
// grid = (16, 32), block = 256.  blockIdx.x = b*4 + feChunk, blockIdx.y = t-chunk.
// partial layout: [32][4][1024]
// ---------------------------------------------------------------------------
__global__ __launch_bounds__(256) void reduce_t_stage1(
    const float* __restrict__ x, float* __restrict__ partial) {
  const int bx  = blockIdx.x;
  const int ty  = blockIdx.y;
  const int tid = threadIdx.x;
  const int b   = bx >> 2;
  const int fe  = (bx & 3) * 256 + tid;
  const float* base = x + ((long)b * Tn + (long)ty * 64) * FE + fe;
  float s = 0.0f;
#pragma unroll 8
  for (int t = 0; t < 64; ++t) s += base[(long)t * FE];
  partial[((long)ty * Bn + b) * FE + fe] = s;
}

// ---------------------------------------------------------------------------
// Stage 1b: finish reduction, write zero-padded xsum [16][1024] (rows 4..15 = 0)
// grid = 64, block = 256
// ---------------------------------------------------------------------------
__global__ __launch_bounds__(256) void reduce_t_stage2(
    const float* __restrict__ partial, float* __restrict__ xsumPad) {
  const int gid = blockIdx.x * 256 + threadIdx.x;  // 0..16383
  const int m  = gid >> 10;
  const int fe = gid & (FE - 1);
  float s = 0.0f;
  if (m < Bn) {
#pragma unroll 8
    for (int ty = 0; ty < 32; ++ty) s += partial[(ty * Bn + m) * FE + fe];
  }
  xsumPad[gid] = s;
}

// ---------------------------------------------------------------------------
// Stage 2: cc = LayerNorm(Wc @ c_flat) * g + b      (Wc: [1024][256])
// one block of 256 threads; each thread owns 4 rows; LDS block reductions.
// ---------------------------------------------------------------------------
__global__ __launch_bounds__(256) void cond_layernorm_kernel(
    const float* __restrict__ c, const float* __restrict__ Wc,
    const float* __restrict__ g, const float* __restrict__ bb,
    float* __restrict__ cc_out) {
  __shared__ float sc[CDIM];
  __shared__ float red[256];
  const int tid = threadIdx.x;
  sc[tid] = c[tid];
  __syncthreads();

  float acc[4];
#pragma unroll
  for (int r = 0; r < 4; ++r) {
    const float* w = Wc + (tid * 4 + r) * CDIM;
    float s = 0.0f;
#pragma unroll 4
    for (int k = 0; k < CDIM; ++k) s += w[k] * sc[k];
    acc[r] = s;
  }
  // mean
  red[tid] = acc[0] + acc[1] + acc[2] + acc[3];
  __syncthreads();
  for (int off = 128; off > 0; off >>= 1) {
    if (tid < off) red[tid] += red[tid + off];
    __syncthreads();
  }
  const float mu = red[0] * (1.0f / FE);
  __syncthreads();
  // variance
  float lv = 0.0f;
#pragma unroll
  for (int r = 0; r < 4; ++r) { float d = acc[r] - mu; lv += d * d; }
  red[tid] = lv;
  __syncthreads();
  for (int off = 128; off > 0; off >>= 1) {
    if (tid < off) red[tid] += red[tid + off];
    __syncthreads();
  }
  const float rstd = rsqrtf(red[0] * (1.0f / FE) + LN_EPS);
#pragma unroll
  for (int r = 0; r < 4; ++r) {
    const int row = tid * 4 + r;
    cc_out[row] = (acc[r] - mu) * rstd * g[row] + bb[row];
  }
}

// ---------------------------------------------------------------------------
// Stage 3: modulate padded activation by cc:  dst[m][k] = src[m][k] * cc[k]
// (rows 4..15 stay zero).  grid = 64, block = 256.
// ---------------------------------------------------------------------------
__global__ __launch_bounds__(256) void modulate_kernel(
    const float* __restrict__ src, const float* __restrict__ cc,
    float* __restrict__ dst) {
  const int gid = blockIdx.x * 256 + threadIdx.x;
  dst[gid] = src[gid] * cc[gid & (FE - 1)];
}

// ---------------------------------------------------------------------------
// Stage 4: WMMA GEMM  D[16][1024] = A[16][1024] @ Wl^T + biasScale*bl
//   D[m][n] = sum_k A[m][k] * Wl[n][k]     (so B-frag[k][n] = Wl[n][k])
// V_WMMA_F32_16X16X4_F32, 256 chained k-steps, one wave32 per 16-col N tile.
// Fragment layouts per ISA 7.12.2:
//   A 16x4 f32: lane(0-15)=M, v0:{K0 | K2 hi-lanes}, v1:{K1 | K3 hi-lanes}
//   B  4x16 f32: lane&15 = N, v0:{K0 | K1 hi-lanes}, v1:{K2 | K3 hi-lanes}
//   C/D: VGPR r -> M=r (lanes 0-15) / M=r+8 (lanes 16-31), N = lane&15
// grid = 64, block = 32 (EXEC all-ones as WMMA requires).
// ---------------------------------------------------------------------------
__global__ __launch_bounds__(32) void wmma_gemm_mod_kernel(
    const float* __restrict__ A, const float* __restrict__ Wl,
    const float* __restrict__ bl, float biasScale,
    float* __restrict__ D) {
  const int lane = threadIdx.x;          // 0..31
  const int nl   = lane & 15;
  const int hi   = lane >> 4;            // 0 | 1
  const int col  = blockIdx.x * 16 + nl; // output column == Wl row
  const float* __restrict__ wrow = Wl + (long)col * FE;  // contiguous in k
  const float* __restrict__ arow = A + (long)nl * FE;    // A row for this lane's M

  v8f acc = {};
#pragma unroll 4
  for (int k4 = 0; k4 < FE; k4 += 4) {
    v2f a = { arow[k4 + 2 * hi], arow[k4 + 2 * hi + 1] };
    v2f b = { wrow[k4 + hi],     wrow[k4 + hi + 2]     };
    acc = __builtin_amdgcn_wmma_f32_16x16x4_f32(
        /*neg_a=*/false, a, /*neg_b=*/false, b,
        /*c_mod=*/(short)0, acc, /*reuse_a=*/false, /*reuse_b=*/false);
  }

  const float bias = biasScale * bl[col];
#pragma unroll
  for (int r = 0; r < 8; ++r) {
    const int m = r + 8 * hi;
    D[m * FE + col] = acc[r] + (m < Bn ? bias : 0.0f);
  }
}

// ---------------------------------------------------------------------------
// Stage 5: broadcast orow[b][fe] to out[b][t][fe] for all t (float4 stores).
// grid = 8192, block = 256  ->  2,097,152 float4 = 33.5 MB.
// ---------------------------------------------------------------------------
__global__ __launch_bounds__(256) void broadcast_kernel(
    const float* __restrict__ orowPad, float* __restrict__ out) {
  const long gid = (long)blockIdx.x * 256 + threadIdx.x;  // float4 index
  const int  feq = (int)(gid & 255);                      // 256 float4 per row
  const long bt  = gid >> 8;                              // b*T + t
  const int  b   = (int)(bt >> 11);                       // T = 2048
  const float4 v = ((const float4*)orowPad)[b * 256 + feq];
  ((float4*)out)[gid] = v;
}

// ---------------------------------------------------------------------------
extern "C" void kernel_launch(void* const* d_in, const int* in_sizes, int n_in,
                              void* d_out, int out_size, void* d_ws, size_t ws_size,
                              hipStream_t stream) {
  (void)in_sizes; (void)n_in; (void)out_size; (void)ws_size;
  const float* x    = (const float*)d_in[0];
  const float* c    = (const float*)d_in[1];
  // d_in[2] (C) and all q_*/k_* params are provably dead: softmax rows sum to 1
  // and the reference einsum sums W over exactly that axis.
  const float* v_Wl = (const float*)d_in[13];
  const float* v_bl = (const float*)d_in[14];
  const float* v_Wc = (const float*)d_in[15];
  const float* v_g  = (const float*)d_in[16];
  const float* v_b  = (const float*)d_in[17];
  const float* o_Wl = (const float*)d_in[18];
  const float* o_bl = (const float*)d_in[19];
  const float* o_Wc = (const float*)d_in[20];
  const float* o_g  = (const float*)d_in[21];
  const float* o_b  = (const float*)d_in[22];

  float* ws        = (float*)d_ws;
  float* partial   = ws;                 // 32*4*1024 = 131072
  float* xsumPad   = ws + 131072;        // 16*1024
  float* ccv       = ws + 147456;        // 1024
  float* cco       = ws + 148480;        // 1024
  float* modA1     = ws + 149504;        // 16*1024
  float* vsumPad   = ws + 165888;        // 16*1024
  float* modA2     = ws + 182272;        // 16*1024
  float* orowPad   = ws + 198656;        // 16*1024

  // xsum[b] = sum_t x[b,t,:]
  reduce_t_stage1<<<dim3(16, 32), 256, 0, stream>>>(x, partial);
  reduce_t_stage2<<<64, 256, 0, stream>>>(partial, xsumPad);
  // conditioning vectors (only v and o survive)
  cond_layernorm_kernel<<<1, 256, 0, stream>>>(c, v_Wc, v_g, v_b, ccv);
  cond_layernorm_kernel<<<1, 256, 0, stream>>>(c, o_Wc, o_g, o_b, cco);
  // vsum[b] = (xsum[b] * ccv) @ v_Wl^T + T * v_bl   (sum_t commutes: v linear in x)
  modulate_kernel<<<64, 256, 0, stream>>>(xsumPad, ccv, modA1);
  wmma_gemm_mod_kernel<<<64, 32, 0, stream>>>(modA1, v_Wl, v_bl, (float)Tn, vsumPad);
  // orow[b] = (vsum[b] * cco) @ o_Wl^T + o_bl
  modulate_kernel<<<64, 256, 0, stream>>>(vsumPad, cco, modA2);
  wmma_gemm_mod_kernel<<<64, 32, 0, stream>>>(modA2, o_Wl, o_bl, 1.0f, orowPad);
  // out[b,t,:] = orow[b,:]  (t-independent)
  broadcast_kernel<<<8192, 256, 0, stream>>>(orowPad, (float*)d_out);
}